// CircleLossWithHardMining_47064251630175
// MI455X (gfx1250) — compile-verified
//
#include <hip/hip_runtime.h>
#include <hip/hip_bf16.h>
#include <math.h>

typedef __attribute__((ext_vector_type(2))) float v2f;
typedef __attribute__((ext_vector_type(8))) float v8f;

#define N_TOT   4096
#define K_DIM   512
#define LDA     516      // 512 + 4 pad -> conflict-free ds_load_b64
#define GAMMA_C 256.0f
#define O_P_C   1.25f
#define O_N_C   (-0.25f)
#define D_P_C   0.75f
#define D_N_C   0.25f
#define NEG_INF_C (-3.0e38f)

// ---------------- Kernel 1: L2-normalize rows, zero global accumulators ---------------
__global__ void __launch_bounds__(256)
normalize_rows_kernel(const float* __restrict__ x, float* __restrict__ xn,
                      float* __restrict__ acc) {
  __shared__ float red[256];
  const int row = blockIdx.x;
  const int tid = threadIdx.x;
  float ss = 0.f;
  for (int c = tid; c < K_DIM; c += 256) { float v = x[row * K_DIM + c]; ss += v * v; }
  red[tid] = ss;
  __syncthreads();
  for (int off = 128; off > 0; off >>= 1) {
    if (tid < off) red[tid] += red[tid + off];
    __syncthreads();
  }
  const float inv = 1.0f / sqrtf(red[0]);
  for (int c = tid; c < K_DIM; c += 256) xn[row * K_DIM + c] = x[row * K_DIM + c] * inv;
  if (row == 0 && tid == 0) { acc[0] = 0.f; acc[1] = 0.f; }
}

// ---------------- Kernel 2: fused WMMA sim + circle-loss row reduction ----------------
// One workgroup = one 16-row strip. 8 waves; wave w covers column-tile groups
// g = w + 8p (p=0..7), each group = 4 tiles of 16 columns sharing one A fetch.
__global__ void __launch_bounds__(256)
circle_loss_kernel(const float* __restrict__ xn, const int* __restrict__ tgt,
                   float* __restrict__ acc) {
  __shared__ __align__(16) float ldsA[16 * LDA];
  __shared__ float stats[8][16][8];   // [wave][row][mp,sp,mn,sn,smin,smax,np,nn]

  const int tid  = threadIdx.x;
  const int wave = tid >> 5;          // wave32
  const int lane = tid & 31;
  const int half = lane >> 4;
  const int ln   = lane & 15;
  const int ibase = blockIdx.x * 16;

  // Stage the 16x512 A strip into LDS (padded row stride).
  for (int idx = tid; idx < 16 * K_DIM; idx += 256) {
    const int r = idx >> 9;           // K_DIM == 512
    const int c = idx & (K_DIM - 1);
    ldsA[r * LDA + c] = xn[(ibase + r) * K_DIM + c];
  }
  __syncthreads();

  // C-matrix layout: VGPR r holds row (r + 8*half), lane holds column ln.
  int irow[8]; int ti[8];
#pragma unroll
  for (int r = 0; r < 8; ++r) { irow[r] = ibase + r + 8 * half; ti[r] = tgt[irow[r]]; }

  // Per-lane streaming state over this lane's subset of columns (8 rows each).
  float mp[8], sp[8], mn_[8], sn[8], smin[8], smax[8];
  int   np[8], nn[8];
#pragma unroll
  for (int r = 0; r < 8; ++r) {
    mp[r] = NEG_INF_C; sp[r] = 0.f; mn_[r] = NEG_INF_C; sn[r] = 0.f;
    smin[r] = 3.0e38f; smax[r] = NEG_INF_C; np[r] = 0; nn[r] = 0;
  }

  for (int p = 0; p < 8; ++p) {
    const int colbase = (wave + 8 * p) * 64;

    // A-layout (16x16x4 f32): lane ln, k-pair base = kk + 2*half.
    const float* aptr = &ldsA[ln * LDA + 2 * half];
    const float* bptr[4];
#pragma unroll
    for (int t = 0; t < 4; ++t) {
      bptr[t] = xn + (size_t)(colbase + t * 16 + ln) * K_DIM + 2 * half;
      __builtin_prefetch(bptr[t], 0, 1);   // global_prefetch_b8 hint into L2
    }

    v8f cacc[4] = {};
    for (int kk = 0; kk < K_DIM; kk += 4) {
      const v2f a = *(const v2f*)(aptr + kk);
#pragma unroll
      for (int t = 0; t < 4; ++t) {
        const v2f b = *(const v2f*)(bptr[t] + kk);
        cacc[t] = __builtin_amdgcn_wmma_f32_16x16x4_f32(
            /*neg_a=*/false, a, /*neg_b=*/false, b,
            /*c_mod=*/(short)0, cacc[t], /*reuse_a=*/false, /*reuse_b=*/false);
      }
    }

    // Stream this group's 4 tiles into the online lse / hard-mining state.
#pragma unroll
    for (int t = 0; t < 4; ++t) {
      const int j  = colbase + t * 16 + ln;
      const int tj = tgt[j];
#pragma unroll
      for (int r = 0; r < 8; ++r) {
        const float s = cacc[t][r];
        const bool same = (tj == ti[r]);
        if (same) {
          if (j != irow[r]) {                       // positive (off-diagonal)
            np[r] += 1;
            smin[r] = fminf(smin[r], s);
            const float al = fmaxf(O_P_C - s, 0.f);
            const float tv = -GAMMA_C * al * (s - D_P_C);
            const float mm = fmaxf(mp[r], tv);
            sp[r] = sp[r] * expf(mp[r] - mm) + expf(tv - mm);
            mp[r] = mm;
          }
        } else {                                    // negative
          nn[r] += 1;
          smax[r] = fmaxf(smax[r], s);
          const float al = fmaxf(s - O_N_C, 0.f);
          const float tv = GAMMA_C * al * (s - D_N_C);
          const float mm = fmaxf(mn_[r], tv);
          sn[r] = sn[r] * expf(mn_[r] - mm) + expf(tv - mm);
          mn_[r] = mm;
        }
      }
    }
  }

  // Reduce across the 16 lanes of each half (rows differ between halves).
#pragma unroll
  for (int r = 0; r < 8; ++r) {
#pragma unroll
    for (int off = 1; off < 16; off <<= 1) {
      float m2, s2, mm;
      m2 = __shfl_xor(mp[r], off, 16);  s2 = __shfl_xor(sp[r], off, 16);
      mm = fmaxf(mp[r], m2); sp[r] = sp[r] * expf(mp[r] - mm) + s2 * expf(m2 - mm); mp[r] = mm;
      m2 = __shfl_xor(mn_[r], off, 16); s2 = __shfl_xor(sn[r], off, 16);
      mm = fmaxf(mn_[r], m2); sn[r] = sn[r] * expf(mn_[r] - mm) + s2 * expf(m2 - mm); mn_[r] = mm;
      smin[r] = fminf(smin[r], __shfl_xor(smin[r], off, 16));
      smax[r] = fmaxf(smax[r], __shfl_xor(smax[r], off, 16));
      np[r] += __shfl_xor(np[r], off, 16);
      nn[r] += __shfl_xor(nn[r], off, 16);
    }
  }

  if (ln == 0) {
#pragma unroll
    for (int r = 0; r < 8; ++r) {
      float* q = &stats[wave][r + 8 * half][0];
      q[0] = mp[r]; q[1] = sp[r]; q[2] = mn_[r]; q[3] = sn[r];
      q[4] = smin[r]; q[5] = smax[r]; q[6] = (float)np[r]; q[7] = (float)nn[r];
    }
  }
  __syncthreads();

  // Final merge across 8 waves + hard-mining correction + softplus.
  if (tid < 16) {
    const int row = tid;
    float Mp = NEG_INF_C, Sp = 0.f, Mn = NEG_INF_C, Sn = 0.f;
    float Smin = 3.0e38f, Smax = NEG_INF_C, Np = 0.f, Nn = 0.f;
    for (int w = 0; w < 8; ++w) {
      const float* q = &stats[w][row][0];
      float m2 = q[0], s2 = q[1];
      float mm = fmaxf(Mp, m2); Sp = Sp * expf(Mp - mm) + s2 * expf(m2 - mm); Mp = mm;
      m2 = q[2]; s2 = q[3];
      mm = fmaxf(Mn, m2); Sn = Sn * expf(Mn - mm) + s2 * expf(m2 - mm); Mn = mm;
      Smin = fminf(Smin, q[4]); Smax = fmaxf(Smax, q[5]); Np += q[6]; Nn += q[7];
    }
    if (Np > 1.5f) {   // hardest positive gets alpha*2: swap its exp term
      const float a1 = fmaxf(O_P_C - Smin, 0.f);
      const float t1 = -GAMMA_C * a1 * (Smin - D_P_C);
      const float t2 = 2.f * t1;
      Sp = Sp - expf(t1 - Mp);
      const float mm = fmaxf(Mp, t2);
      Sp = Sp * expf(Mp - mm) + expf(t2 - mm);
      Mp = mm;
    }
    if (Nn > 1.5f) {   // hardest negative
      const float a1 = fmaxf(Smax - O_N_C, 0.f);
      const float t1 = GAMMA_C * a1 * (Smax - D_N_C);
      const float t2 = 2.f * t1;
      Sn = Sn - expf(t1 - Mn);
      const float mm = fmaxf(Mn, t2);
      Sn = Sn * expf(Mn - mm) + expf(t2 - mm);
      Mn = mm;
    }
    if (Np > 0.5f && Nn > 0.5f) {
      Sp = fmaxf(Sp, 1e-30f);
      Sn = fmaxf(Sn, 1e-30f);
      const float z = (Mp + logf(Sp)) + (Mn + logf(Sn));
      const float loss = fmaxf(z, 0.f) + log1pf(expf(-fabsf(z)));  // stable softplus
      atomicAdd(&acc[0], loss);
      atomicAdd(&acc[1], 1.0f);
    }
  }
}

// ---------------- Kernel 3: finalize ----------------
__global__ void finalize_kernel(const float* __restrict__ acc, float* __restrict__ out) {
  const float nv = fmaxf(acc[1], 1.0f);
  out[0] = acc[0] / nv;
}

extern "C" void kernel_launch(void* const* d_in, const int* in_sizes, int n_in,
                              void* d_out, int out_size, void* d_ws, size_t ws_size,
                              hipStream_t stream) {
  const float* inputs  = (const float*)d_in[0];  // [4096,512] f32
  const int*   targets = (const int*)d_in[1];    // [4096] int
  float* out = (float*)d_out;

  float* xn  = (float*)d_ws;                     // normalized inputs: 4096*512 f32
  float* acc = xn + (size_t)N_TOT * K_DIM;       // [sum_loss, n_valid]

  normalize_rows_kernel<<<N_TOT, 256, 0, stream>>>(inputs, xn, acc);
  circle_loss_kernel<<<N_TOT / 16, 256, 0, stream>>>(xn, targets, acc);
  finalize_kernel<<<1, 1, 0, stream>>>(acc, out);
}